// GumbelMarkovBLSTM_3607772528772
// MI455X (gfx1250) — compile-verified
//
#include <hip/hip_runtime.h>

// GumbelMarkovBLSTM for MI455X (gfx1250, wave32).
// Pipeline:
//   k_pad        : zero-pad / transpose weight panels into WMMA-friendly layouts
//   k_precompute : fused wi/lz GEMMs via v_wmma_f32_16x16x4_f32 (reads x once,
//                  all loads unconditional & coalesced -> no exec-mask branches)
//   k_scan       : serial T-loop, 1 workgroup per batch, LDS-resident W_hh/trans,
//                  shfl_xor + 2-wave LDS combine softmax; branch-free prefetch
//   k_decode     : out_logit GEMM via v_wmma_f32_16x16x4_f32 on padded panels
// Workspace: wi(B*T*49) + lz(B*T*49) + enc(B*T*52) + Wpad(512*64) + Fpad(512*64)
//            + Dpad(52*80) floats  ~= 78.9 MB.

typedef float v2f __attribute__((ext_vector_type(2)));
typedef float v8f __attribute__((ext_vector_type(8)));

#define Bsz   64
#define Tlen  2048
#define Dsz   512
#define Ksz   49
#define Csz   65
#define KPAD  64     // N padding for the wi/lz GEMMs
#define KP52  52     // K padding for the decode GEMM (enc row stride)
#define CPAD  80     // C padding for the decode GEMM

// ---------------------------------------------------------------------------
// Kernel 0: build padded/transposed weight panels.
//   Wpad[d*64+n] = (n<49) ? W_ih[d*49+n]  : 0       (512 x 64)
//   Fpad[d*64+n] = (n<49) ? fc_W[n*512+d] : 0       (512 x 64)
//   Dpad[k*80+c] = (k<49 && c<65) ? dec_W[c*49+k] : 0   (52 x 80)
// ---------------------------------------------------------------------------
__global__ __launch_bounds__(256) void k_pad(
    const float* __restrict__ W_ih, const float* __restrict__ fc_W,
    const float* __restrict__ dec_W,
    float* __restrict__ Wpad, float* __restrict__ Fpad, float* __restrict__ Dpad)
{
  const int i = blockIdx.x * 256 + threadIdx.x;
  const int NW = Dsz * KPAD;             // 32768
  const int ND = KP52 * CPAD;            // 4160
  if (i < NW) {
    const int d = i >> 6, n = i & 63;
    Wpad[i] = (n < Ksz) ? W_ih[d * Ksz + n] : 0.f;
    Fpad[i] = (n < Ksz) ? fc_W[(size_t)n * Dsz + d] : 0.f;
  } else {
    const int j = i - NW;
    if (j < ND) {
      const int k = j / CPAD, c = j % CPAD;
      Dpad[j] = (k < Ksz && c < Csz) ? dec_W[c * Ksz + k] : 0.f;
    }
  }
}

// ---------------------------------------------------------------------------
// Kernel 1: wi[b,t,k] = sum_d x[b,d,t]*W_ih[d,k] + bias[k]
//           lz[b,t,k] = sum_d x[b,d,t]*fc_W[k,d] + fc_b[k]
// Workgroup = 16 consecutive t of one batch; 4 waves, wave w owns N-tile
// columns [16w,16w+16); K loop over D in steps of 4; 2 WMMAs per step.
// WMMA f32 16x16x4 lane layout:
//   A (16x4): lanes 0-15 hold M=lane, {v0,v1}={K0,K1}; lanes 16-31 {K2,K3}
//   B (4x16): lanes 0-15 hold N=lane, {v0,v1}={K0,K1}; lanes 16-31 {K2,K3}
//   C/D:      vgpr r: lanes 0-15 -> M=r, N=lane; lanes 16-31 -> M=r+8, N=lane-16
// ---------------------------------------------------------------------------
__global__ __launch_bounds__(128) void k_precompute(
    const float* __restrict__ x,      // (B, D, T)
    const float* __restrict__ Wpad,   // (D, 64) padded
    const float* __restrict__ Fpad,   // (D, 64) padded (fc_W transposed)
    const float* __restrict__ bias,   // (K)
    const float* __restrict__ fc_b,   // (K)
    float* __restrict__ wi_ws,        // (B*T, 49)
    float* __restrict__ lz_ws)        // (B*T, 49)
{
  const int tid   = threadIdx.x;
  const int lane  = tid & 31;
  const int wave  = tid >> 5;
  const int bt0   = blockIdx.x * 16;          // T divisible by 16 -> no batch crossing
  const int b     = bt0 / Tlen;
  const int t0    = bt0 % Tlen;
  const int mrow  = lane & 15;
  const int khalf = (lane >> 4) << 1;         // 0 for lanes 0-15, 2 for lanes 16-31
  const int n     = wave * 16 + mrow;         // 0..63, always in-bounds of padded panels

  const float* xb = x + (size_t)b * Dsz * Tlen + t0 + mrow;
  const float* wp = Wpad + n;
  const float* fp = Fpad + n;

  v8f cwi = {0.f, 0.f, 0.f, 0.f, 0.f, 0.f, 0.f, 0.f};
  v8f clz = {0.f, 0.f, 0.f, 0.f, 0.f, 0.f, 0.f, 0.f};

  for (int d0 = 0; d0 < Dsz; d0 += 4) {
    const int d = d0 + khalf;
    v2f a, bwi, blz;
    a.x   = xb[(size_t)d * Tlen];             // coalesced: 16 consecutive t per half-wave
    a.y   = xb[(size_t)(d + 1) * Tlen];
    bwi.x = wp[d * KPAD];                     // coalesced 64B row reads, no guards
    bwi.y = wp[(d + 1) * KPAD];
    blz.x = fp[d * KPAD];
    blz.y = fp[(d + 1) * KPAD];
    cwi = __builtin_amdgcn_wmma_f32_16x16x4_f32(false, a, false, bwi, (short)0, cwi, false, false);
    clz = __builtin_amdgcn_wmma_f32_16x16x4_f32(false, a, false, blz, (short)0, clz, false, false);
  }

  if (n < Ksz) {
    const float bw = bias[n];
    const float bl = fc_b[n];
#pragma unroll
    for (int r = 0; r < 8; ++r) {
      const int m = (lane < 16) ? r : (r + 8);
      const size_t idx = (size_t)(bt0 + m) * Ksz + n;
      wi_ws[idx] = cwi[r] + bw;
      lz_ws[idx] = clz[r] + bl;
    }
  }
}

// ---------------------------------------------------------------------------
// Kernel 2: the serial scan. One workgroup (64 threads = 2 wave32s) per batch.
// Thread k (<49) owns state element k. W_hh/trans live in LDS; z carried in LDS.
// Per-step global reads are unconditional (clamped index) and prefetched one
// iteration ahead. enc is written with row stride 52, rows 49..51 zeroed, so
// the decode kernel can load A fragments guard-free.
// ---------------------------------------------------------------------------
__global__ __launch_bounds__(64) void k_scan(
    const float* __restrict__ gumbel,  // (B, T, 49)
    const float* __restrict__ W_hh,    // (49, 49)
    const float* __restrict__ trans,   // (49, 49)
    const float* __restrict__ wi_ws,   // (B*T, 49)
    const float* __restrict__ lz_ws,   // (B*T, 49)
    float* __restrict__ in_logit,      // (B, T, 49)  -> d_out part 1
    float* __restrict__ enc_ws)        // (B*T, 52) zero-padded
{
  __shared__ float whh[Ksz * Ksz];
  __shared__ float trs[Ksz * Ksz];
  __shared__ float zsh[64];
  __shared__ float red[2];

  const int tid    = threadIdx.x;
  const int b      = blockIdx.x;
  const bool valid = (tid < Ksz);
  const int kk     = valid ? tid : (Ksz - 1);  // clamp: dead lanes mirror k=48

  for (int i = tid; i < Ksz * Ksz; i += 64) {
    whh[i] = W_hh[i];
    trs[i] = trans[i];
  }
  zsh[tid] = 0.f;
  __syncthreads();

  const float* wi_p = wi_ws  + (size_t)b * Tlen * Ksz + kk;
  const float* lz_p = lz_ws  + (size_t)b * Tlen * Ksz + kk;
  const float* gm_p = gumbel + (size_t)b * Tlen * Ksz + kk;

  // unconditional preload of t=0 (dead lanes duplicate element 48: finite, masked later)
  float wi_c = wi_p[0];
  float lz_c = lz_p[0];
  float g_c  = gm_p[0];

  for (int t = 0; t < Tlen; ++t) {
    // branch-free software prefetch of step t+1 (clamped at the last step)
    const int tn = (t + 1 < Tlen) ? (t + 1) : t;
    const float wi_n = wi_p[tn * Ksz];
    const float lz_n = lz_p[tn * Ksz];
    const float g_n  = gm_p[tn * Ksz];

    // g[k] = wi + sum_j z[j]*W_hh[j,k]; prior[k] = sum_j z[j]*trans[j,k]
    float accg = wi_c;
    float accp = 0.f;
#pragma unroll
    for (int j = 0; j < Ksz; ++j) {
      const float zj = zsh[j];                  // broadcast read
      accg = fmaf(zj, whh[j * Ksz + kk], accg); // consecutive banks across lanes
      accp = fmaf(zj, trs[j * Ksz + kk], accp);
    }

    const float s     = 1.f / (1.f + __expf(-accg));
    const float logit = s * accp + (1.f - s) * lz_c;
    float y = valid ? (logit + g_c) : -3.4e38f; // TEMP == 1.0

    // softmax over 49 values spread across 2 wave32s
    float m = y;
#pragma unroll
    for (int off = 16; off > 0; off >>= 1) m = fmaxf(m, __shfl_xor(m, off, 32));
    if ((tid & 31) == 0) red[tid >> 5] = m;
    __syncthreads();                            // also fences matvec reads of zsh
    m = fmaxf(red[0], red[1]);
    __syncthreads();

    const float e = valid ? __expf(y - m) : 0.f;
    float ssum = e;
#pragma unroll
    for (int off = 16; off > 0; off >>= 1) ssum += __shfl_xor(ssum, off, 32);
    if ((tid & 31) == 0) red[tid >> 5] = ssum;
    __syncthreads();
    ssum = red[0] + red[1];

    const float z1 = e / ssum;                  // exactly 0 for lanes >= 49
    zsh[tid] = z1;                              // state for step t+1
    if (valid)
      in_logit[(size_t)b * Tlen * Ksz + (size_t)t * Ksz + tid] = logit;
    if (tid < KP52)                             // rows 49..51 get zeros (padding)
      enc_ws[((size_t)b * Tlen + t) * KP52 + tid] = z1;
    __syncthreads();                            // publish zsh before next matvec

    wi_c = wi_n; lz_c = lz_n; g_c = g_n;
  }
}

// ---------------------------------------------------------------------------
// Kernel 3: out_logit[bt, c] = sum_k enc[bt,k]*dec_W[c,k] + dec_b[c]
// One wave per 16x16 output tile; K padded 49->52 (13 WMMA steps, guard-free),
// C padded 65->80 (grid.y = 5 N-tiles). A loads are contiguous pairs (b64),
// B loads are coalesced rows of the k-major Dpad panel.
// ---------------------------------------------------------------------------
__global__ __launch_bounds__(32) void k_decode(
    const float* __restrict__ enc_ws,  // (B*T, 52) zero-padded
    const float* __restrict__ Dpad,    // (52, 80) padded, k-major
    const float* __restrict__ dec_b,   // (65)
    float* __restrict__ out_logit)     // (B*T, 65)  -> d_out part 2
{
  const int lane  = threadIdx.x;
  const int bt0   = blockIdx.x * 16;
  const int c0    = blockIdx.y * 16;
  const int mrow  = lane & 15;
  const int khalf = (lane >> 4) << 1;
  const int c     = c0 + mrow;                 // 0..79, in-bounds of padded panel

  const float* arow = enc_ws + (size_t)(bt0 + mrow) * KP52;
  const float* brow = Dpad + c;

  v8f acc = {0.f, 0.f, 0.f, 0.f, 0.f, 0.f, 0.f, 0.f};

  for (int k0 = 0; k0 < KP52; k0 += 4) {       // 13 iterations
    const int k = k0 + khalf;                  // k, k+1 <= 51: always in-bounds
    v2f a, bm;
    a.x  = arow[k];                            // contiguous pair -> b64-mergeable
    a.y  = arow[k + 1];
    bm.x = brow[k * CPAD];                     // coalesced 64B row reads
    bm.y = brow[(k + 1) * CPAD];
    acc = __builtin_amdgcn_wmma_f32_16x16x4_f32(false, a, false, bm, (short)0, acc, false, false);
  }

  if (c < Csz) {
    const float bb = dec_b[c];
#pragma unroll
    for (int r = 0; r < 8; ++r) {
      const int m = (lane < 16) ? r : (r + 8);
      out_logit[(size_t)(bt0 + m) * Csz + c] = acc[r] + bb;
    }
  }
}

// ---------------------------------------------------------------------------
extern "C" void kernel_launch(void* const* d_in, const int* in_sizes, int n_in,
                              void* d_out, int out_size, void* d_ws, size_t ws_size,
                              hipStream_t stream) {
  const float* x      = (const float*)d_in[0];  // (B, D, T)
  const float* gumbel = (const float*)d_in[1];  // (B, T, K)
  const float* fc_W   = (const float*)d_in[2];  // (K, D)
  const float* fc_b   = (const float*)d_in[3];  // (K)
  const float* W_ih   = (const float*)d_in[4];  // (D, K)
  const float* W_hh   = (const float*)d_in[5];  // (K, K)
  const float* bias   = (const float*)d_in[6];  // (K)
  const float* trans  = (const float*)d_in[7];  // (K, K)
  const float* dec_W  = (const float*)d_in[8];  // (C, K)
  const float* dec_b  = (const float*)d_in[9];  // (C)

  const size_t NK  = (size_t)Bsz * Tlen * Ksz;  // 6,422,528
  const size_t NE  = (size_t)Bsz * Tlen * KP52; // 6,815,744

  float* in_logit  = (float*)d_out;             // B*T*49
  float* out_logit = (float*)d_out + NK;        // B*T*65

  float* wi_ws  = (float*)d_ws;                 // B*T*49
  float* lz_ws  = wi_ws + NK;                   // B*T*49
  float* enc_ws = lz_ws + NK;                   // B*T*52
  float* Wpad   = enc_ws + NE;                  // 512*64
  float* Fpad   = Wpad + Dsz * KPAD;            // 512*64
  float* Dpad   = Fpad + Dsz * KPAD;            // 52*80

  const int padN = Dsz * KPAD + KP52 * CPAD;    // 36928 elements to fill
  k_pad<<<(padN + 255) / 256, 256, 0, stream>>>(W_ih, fc_W, dec_W, Wpad, Fpad, Dpad);
  k_precompute<<<(Bsz * Tlen) / 16, 128, 0, stream>>>(x, Wpad, Fpad, bias, fc_b,
                                                      wi_ws, lz_ws);
  k_scan<<<Bsz, 64, 0, stream>>>(gumbel, W_hh, trans, wi_ws, lz_ws,
                                 in_logit, enc_ws);
  k_decode<<<dim3((Bsz * Tlen) / 16, 5), 32, 0, stream>>>(enc_ws, Dpad, dec_b,
                                                          out_logit);
}